// Symmetry_Set_Basis_27255862460523
// MI455X (gfx1250) — compile-verified
//
#include <hip/hip_runtime.h>

// B=2048 rows, N=4096 set elements, NODE=7 features.
#define N_SET 4096
#define NODE 7
#define BLOCK 256
#define WAVES (BLOCK / 32)

typedef __attribute__((ext_vector_type(2))) float v2f;
typedef __attribute__((ext_vector_type(8))) float v8f;

__device__ __forceinline__ float sgnf(float x) {
    return (x > 0.f) ? 1.f : ((x < 0.f) ? -1.f : 0.f);
}
__device__ __forceinline__ float sroot2(float x) { return sgnf(x) * sqrtf(fabsf(x)); }
__device__ __forceinline__ float sroot4(float x) { return sgnf(x) * sqrtf(sqrtf(fabsf(x))); }

__global__ __launch_bounds__(BLOCK)
void symset_kernel(const float* __restrict__ q, const float* __restrict__ v,
                   const float* __restrict__ wq_w, const float* __restrict__ wq_b,
                   const float* __restrict__ wk_w, const float* __restrict__ wk_b,
                   float* __restrict__ out) {
    __shared__ float qbuf[N_SET];          // 16 KB: stage q so the epilogue hits LDS, not HBM
    __shared__ float psums[WAVES * 6];

    const int b    = blockIdx.x;
    const int tid  = threadIdx.x;
    const int lane = tid & 31;
    const int wave = tid >> 5;

    const float4* q4  = (const float4*)(q + (size_t)b * N_SET);
    const float4* v4  = (const float4*)(v + (size_t)b * N_SET);
    float4*       o4  = (float4*)(out + (size_t)b * N_SET);
    float4*       qb4 = (float4*)qbuf;

    // ---- Phase 1: stream v and q, accumulate power sums + q moments ----
    float p1 = 0.f, p2 = 0.f, p3 = 0.f, p4 = 0.f, s1 = 0.f, s2 = 0.f;
#pragma unroll
    for (int i = 0; i < N_SET / 4 / BLOCK; ++i) {
        int idx = tid + i * BLOCK;
        float4 vv = v4[idx];
        float4 qq = q4[idx];
        qb4[idx] = qq;
        float x2 = vv.x * vv.x, y2 = vv.y * vv.y, z2 = vv.z * vv.z, w2 = vv.w * vv.w;
        p1 += vv.x + vv.y + vv.z + vv.w;
        p2 += x2 + y2 + z2 + w2;
        p3 += x2 * vv.x + y2 * vv.y + z2 * vv.z + w2 * vv.w;
        p4 += x2 * x2 + y2 * y2 + z2 * z2 + w2 * w2;
        s1 += qq.x + qq.y + qq.z + qq.w;
        s2 += qq.x * qq.x + qq.y * qq.y + qq.z * qq.z + qq.w * qq.w;
    }
    // wave32 reduction
#pragma unroll
    for (int off = 16; off >= 1; off >>= 1) {
        p1 += __shfl_xor(p1, off, 32);
        p2 += __shfl_xor(p2, off, 32);
        p3 += __shfl_xor(p3, off, 32);
        p4 += __shfl_xor(p4, off, 32);
        s1 += __shfl_xor(s1, off, 32);
        s2 += __shfl_xor(s2, off, 32);
    }
    if (lane == 0) {
        psums[wave * 6 + 0] = p1; psums[wave * 6 + 1] = p2; psums[wave * 6 + 2] = p3;
        psums[wave * 6 + 3] = p4; psums[wave * 6 + 4] = s1; psums[wave * 6 + 5] = s2;
    }
    __syncthreads();
    p1 = p2 = p3 = p4 = s1 = s2 = 0.f;
#pragma unroll
    for (int w = 0; w < WAVES; ++w) {
        p1 += psums[w * 6 + 0]; p2 += psums[w * 6 + 1]; p3 += psums[w * 6 + 2];
        p4 += psums[w * 6 + 3]; s1 += psums[w * 6 + 4]; s2 += psums[w * 6 + 5];
    }

    // ---- Phase 2: Newton's identities -> f[7]; g = f@Wk + bk via WMMA ----
    float e1 = p1;
    float e2 = (p1 * p1 - p2) * 0.5f;
    float e3 = (p1 * p1 * p1 - p3 - 3.f * (p2 * p1 - p3)) * (1.f / 6.f);
    float e4 = (e3 * p1 - e2 * p2 + e1 * p3 - p4) * 0.25f;
    float f0 = e1,          f1 = sroot2(e2), f2 = cbrtf(e3), f3 = sroot4(e4);
    float f4v = sroot2(p2), f5 = cbrtf(p3),  f6 = sroot4(p4);

    // A (16x4 f32): lane0 holds K0/K1, lane16 holds K2/K3 for row M=0; rest zero.
    // B (4x16 f32): VGPR0 = rows K0 (lanes 0-15) / K2 (lanes 16-31), VGPR1 = K1 / K3.
    v2f A1 = (v2f){0.f, 0.f}, B1 = (v2f){0.f, 0.f};
    v2f A2 = (v2f){0.f, 0.f}, B2 = (v2f){0.f, 0.f};
    if (lane == 0)  { A1.x = f0; A1.y = f1; A2.x = f4v; A2.y = f5; }
    if (lane == 16) { A1.x = f2; A1.y = f3; A2.x = f6;  A2.y = 0.f; }
    const int  col = lane & 15;
    const bool lo  = lane < 16;
    if (col < NODE) {
        if (lo) {
            B1.x = wk_w[0 * NODE + col];
            B1.y = wk_w[1 * NODE + col];
            B2.x = wk_w[4 * NODE + col];
            B2.y = wk_w[5 * NODE + col];
        } else {
            B1.x = wk_w[2 * NODE + col];
            B1.y = wk_w[3 * NODE + col];
            B2.x = wk_w[6 * NODE + col];
            B2.y = 0.f;
        }
    }
    v8f c = (v8f){0.f, 0.f, 0.f, 0.f, 0.f, 0.f, 0.f, 0.f};
    // K = 0..3
    c = __builtin_amdgcn_wmma_f32_16x16x4_f32(false, A1, false, B1, (short)0, c, false, false);
    // K = 4..6 (padded), accumulating
    c = __builtin_amdgcn_wmma_f32_16x16x4_f32(false, A2, false, B2, (short)0, c, false, false);

    // D row M=0 lives in c[0] of lanes 0..15 (N = lane).
    float alpha_t = 0.f, beta_t = 0.f;
    if (lo && col < NODE) {
        float g   = c[0] + wk_b[col];
        float kn  = g / (sqrtf((float)N_SET) * fabsf(g) + 1e-10f);
        float fj  = (col == 0) ? f0 : (col == 1) ? f1 : (col == 2) ? f2 :
                    (col == 3) ? f3 : (col == 4) ? f4v : (col == 5) ? f5 : f6;
        float wj  = wq_w[col];
        float bj  = wq_b[col];
        float nq  = sqrtf(wj * wj * s2 + 2.f * wj * bj * s1 + (float)N_SET * bj * bj) + 1e-10f;
        float t   = kn * fj / nq;
        alpha_t = wj * t;
        beta_t  = bj * t;
    }
#pragma unroll
    for (int off = 16; off >= 1; off >>= 1) {
        alpha_t += __shfl_xor(alpha_t, off, 32);
        beta_t  += __shfl_xor(beta_t, off, 32);
    }
    const float alpha = alpha_t, beta = beta_t;

    // ---- Phase 3: out = alpha * q + beta (q replayed from LDS) ----
#pragma unroll
    for (int i = 0; i < N_SET / 4 / BLOCK; ++i) {
        int idx = tid + i * BLOCK;
        float4 qq = qb4[idx];
        float4 o;
        o.x = alpha * qq.x + beta;
        o.y = alpha * qq.y + beta;
        o.z = alpha * qq.z + beta;
        o.w = alpha * qq.w + beta;
        o4[idx] = o;
    }
}

extern "C" void kernel_launch(void* const* d_in, const int* in_sizes, int n_in,
                              void* d_out, int out_size, void* d_ws, size_t ws_size,
                              hipStream_t stream) {
    (void)n_in; (void)out_size; (void)d_ws; (void)ws_size;
    const float* q    = (const float*)d_in[0];
    // d_in[1] = k is unused by the reference layer
    const float* v    = (const float*)d_in[2];
    const float* wq_w = (const float*)d_in[3];
    const float* wq_b = (const float*)d_in[4];
    const float* wk_w = (const float*)d_in[5];
    const float* wk_b = (const float*)d_in[6];
    float* out = (float*)d_out;

    const int B = in_sizes[0] / N_SET;  // 2048
    symset_kernel<<<dim3(B), dim3(BLOCK), 0, stream>>>(q, v, wq_w, wq_b, wk_w, wk_b, out);
}